// ContinuousOutputGenerator_90417651515908
// MI455X (gfx1250) — compile-verified
//
#include <hip/hip_runtime.h>
#include <hip/hip_bf16.h>

// ---------------------------------------------------------------------------
// Problem constants (fixed by the reference)
// ---------------------------------------------------------------------------
#define B_DIM 8
#define N_DIM 4096
#define D_DIM 256
#define H_DIM 512
#define O_DIM 256
#define G_DIM 4096     // 64x64 grid
#define GRID_1D 64
#define INV_BW 10.0f   // 1/KERNEL_BW
#define EPS_RBF 1e-8f

// padded LDS strides (stride % 64 == 0 would serialize column reads on banks)
#define LDX (D_DIM + 4)   // 260
#define LDH (H_DIM + 4)   // 516
#define NKC 128           // entity chunk for pooling
#define NCHUNK (N_DIM / NKC)
#define LDW (NKC + 4)     // 132  (w tile stride)
#define LDP (O_DIM + 8)   // 264  (TDM-padded proj tile stride; rows k,k+2 are 16 banks apart)
#define PBUF (NKC * LDP)  // floats per proj LDS buffer (135168 B)
#define PCHUNK_ELTS (NKC * O_DIM)  // 32768 fp32 elements per TDM transfer

typedef __attribute__((ext_vector_type(2))) float v2f;
typedef __attribute__((ext_vector_type(8))) float v8f;
typedef __attribute__((ext_vector_type(4))) unsigned int v4u;
typedef __attribute__((ext_vector_type(8))) int v8i;
typedef __attribute__((ext_vector_type(4))) int v4i;

// D = A(16x4,f32) * B(4x16,f32) + C(16x16,f32), wave32, fp32 matrix pipe
__device__ __forceinline__ v8f wmma_f32_4(v2f a, v2f b, v8f c) {
  return __builtin_amdgcn_wmma_f32_16x16x4_f32(
      /*neg_a=*/false, a, /*neg_b=*/false, b,
      /*c_mod=*/(short)0, c, /*reuse_a=*/false, /*reuse_b=*/false);
}

__device__ __forceinline__ float gelu_exact(float x) {
  return 0.5f * x * (1.0f + erff(x * 0.70710678118654752f));
}

// ---------------------------------------------------------------------------
// TDM: DMA one contiguous proj chunk (128 rows x 256 f32 = 128 KB) into LDS,
// with hardware padding of 8 DWORDs after every 256 DWORDs so the LDS tile
// has a 264-float row stride (bank-conflict-free B-fragment reads).
// Descriptor per CDNA5 ISA ch.8 (D# groups). Issued by one wave; tracked
// with TENSORcnt. Uses the 6-arg builtin form (clang-23 / therock-10.0).
// ---------------------------------------------------------------------------
__device__ __forceinline__ void tdm_load_proj_chunk(const float* gsrc,
                                                    unsigned int lds_byte_off) {
  const unsigned long long ga = (unsigned long long)(uintptr_t)gsrc;

  v4u g0;
  g0.x = 1u;                                   // count=1 (valid user D#)
  g0.y = lds_byte_off;                         // lds_addr [63:32]
  g0.z = (unsigned int)ga;                     // global_addr[31:0]  -> bits 95:64
  g0.w = (unsigned int)((ga >> 32) & 0x01ffffffu) | (2u << 30);  // addr[56:32] | type=2

  v8i g1;
  g1[0] = (2 << 16)        // data_size = 2 -> 4 bytes
        | (1 << 20)        // pad_enable
        | (7 << 22)        // pad_interval = 7 -> every 256 DWORDs (one 256-f32 row)
        | (7 << 25);       // pad_amount  = 7 -> 8 DWORDs pad  => stride 264 f32
  g1[1] = (int)((PCHUNK_ELTS & 0xffffu) << 16);             // tensor_dim0[15:0]
  g1[2] = (int)((PCHUNK_ELTS >> 16) & 0xffffu) | (1 << 16); // tensor_dim0[31:16] | tensor_dim1=1
  g1[3] = (int)((PCHUNK_ELTS & 0xffffu) << 16);             // tensor_dim1 hi(0) | tile_dim0
  g1[4] = 1;                                                // tile_dim1=1, tile_dim2=0
  g1[5] = PCHUNK_ELTS;                                      // tensor_dim0_stride lo
  g1[6] = 0;
  g1[7] = 0;

  v4i g2;
  g2[0] = 1;            // tensor_dim2
  g2[1] = 1;            // tensor_dim3
  g2[2] = PCHUNK_ELTS;  // tensor_dim2_stride lo
  g2[3] = 0;            // stride hi | tile_dim3=0

  v4i g3;
  g3[0] = 0; g3[1] = 0; g3[2] = 0; g3[3] = 0;

  v8i g4;               // extra operand of the 6-arg builtin; zero-filled
  g4[0] = 0; g4[1] = 0; g4[2] = 0; g4[3] = 0;
  g4[4] = 0; g4[5] = 0; g4[6] = 0; g4[7] = 0;

  __builtin_amdgcn_tensor_load_to_lds(g0, g1, g2, g3, g4, /*cpol=*/0);
}

// ---------------------------------------------------------------------------
// Kernel 1: projected = GELU(X @ W1 + b1) @ W2 + b2
//   one block (256 threads = 8 waves) per 16-row strip of X
// ---------------------------------------------------------------------------
__global__ __launch_bounds__(256)
void cog_mlp_kernel(const float* __restrict__ X,
                    const float* __restrict__ W1,
                    const float* __restrict__ b1,
                    const float* __restrict__ W2,
                    const float* __restrict__ b2,
                    float* __restrict__ proj) {
  __shared__ float Xs[16 * LDX];
  __shared__ float Hs[16 * LDH];

  const int tid   = threadIdx.x;
  const int wave  = tid >> 5;
  const int lane  = tid & 31;
  const int mrow  = lane & 15;   // M row of A-frag / N col of B-frag
  const int khalf = lane >> 4;   // which K pair (0: K=0,1  1: K=2,3)
  const long row0 = (long)blockIdx.x * 16;

  // ---- stage X strip [16 x 256] into LDS (coalesced) ----
  for (int i = tid; i < 16 * D_DIM; i += 256) {
    const int m = i >> 8;
    const int k = i & (D_DIM - 1);
    Xs[m * LDX + k] = X[(row0 + m) * D_DIM + k];
  }
  __syncthreads();

  // ---- GEMM1 + GELU: H strip [16 x 512]; 32 col-tiles, 4 per wave ----
  for (int ct = wave; ct < H_DIM / 16; ct += 8) {
    const int col = ct * 16 + mrow;
    const float bias = b1[col];
    v8f acc;
#pragma unroll
    for (int r = 0; r < 8; ++r) acc[r] = bias;

    for (int k = 0; k < D_DIM; k += 4) {
      const int kk = k + 2 * khalf;
      v2f a, bf;
      a.x  = Xs[mrow * LDX + kk];
      a.y  = Xs[mrow * LDX + kk + 1];
      bf.x = W1[(long)kk * H_DIM + col];
      bf.y = W1[(long)(kk + 1) * H_DIM + col];
      acc = wmma_f32_4(a, bf, acc);
    }
#pragma unroll
    for (int r = 0; r < 8; ++r) {
      const int M = r + 8 * khalf;
      Hs[M * LDH + col] = gelu_exact(acc[r]);
    }
  }
  __syncthreads();

  // ---- GEMM2: projected strip [16 x 256]; 16 col-tiles, 2 per wave ----
  for (int ct = wave; ct < O_DIM / 16; ct += 8) {
    const int col = ct * 16 + mrow;
    const float bias = b2[col];
    v8f acc;
#pragma unroll
    for (int r = 0; r < 8; ++r) acc[r] = bias;

    for (int k = 0; k < H_DIM; k += 4) {
      const int kk = k + 2 * khalf;
      v2f a, bf;
      a.x  = Hs[mrow * LDH + kk];
      a.y  = Hs[mrow * LDH + kk + 1];
      bf.x = W2[(long)kk * O_DIM + col];
      bf.y = W2[(long)(kk + 1) * O_DIM + col];
      acc = wmma_f32_4(a, bf, acc);
    }
#pragma unroll
    for (int r = 0; r < 8; ++r) {
      const int M = r + 8 * khalf;
      proj[(row0 + M) * O_DIM + col] = acc[r];
    }
  }
}

// ---------------------------------------------------------------------------
// Kernel 2: out[b,g,o] = (1/(sum_n w + eps)) * sum_n w[g,n] * proj[b,n,o]
//   w[g,n] = exp(-|coord_g - pos_n|^2 / 0.1), computed on the fly into LDS.
//   proj chunks are staged into double-buffered LDS by the Tensor Data Mover
//   (wave 0 issues tensor_load_to_lds for chunk c+1 while all waves run WMMA
//   on chunk c; completion enforced with s_wait_tensorcnt + barrier).
//   grid: (G/16, B); block: 256 threads = 8 waves, each wave owns 2 o-tiles
// ---------------------------------------------------------------------------
__global__ __launch_bounds__(256)
void cog_pool_kernel(const float* __restrict__ pos,
                     const float* __restrict__ proj,
                     float* __restrict__ out) {
  extern __shared__ float smem[];
  float* Wls    = smem;                 // 16 * LDW
  float* rowsum = smem + 16 * LDW;      // 16
  float* Pls    = rowsum + 16;          // 2 * PBUF (16B-aligned: 8512 B offset)

  const int tid   = threadIdx.x;
  const int wave  = tid >> 5;
  const int lane  = tid & 31;
  const int mrow  = lane & 15;
  const int khalf = lane >> 4;
  const int b     = blockIdx.y;
  const int g0    = blockIdx.x * 16;

  if (tid < 16) rowsum[tid] = 0.0f;

  const long pbase_b = (long)b * N_DIM * O_DIM;   // proj[b,0,0]

  // kick off DMA of chunk 0 into buffer 0
  if (wave == 0) {
    tdm_load_proj_chunk(proj + pbase_b,
                        (unsigned int)(uintptr_t)(void*)Pls);
  }
  __syncthreads();   // rowsum init visible (TDM completion handled below)

  const int col0 = (wave * 2) * 16 + mrow;       // o-tile 2*wave
  const int col1 = (wave * 2 + 1) * 16 + mrow;   // o-tile 2*wave+1
  v8f acc0 = {};
  v8f acc1 = {};

  const int   nn     = tid & (NKC - 1);  // this thread's n within the chunk
  const int   grbase = tid >> 7;         // 0 or 1
  const float step   = 2.0f / (GRID_1D - 1);

  for (int c = 0; c < NCHUNK; ++c) {
    const int n0 = c * NKC;

    // ---- phase A: RBF weights for 16 grid rows x NKC entities ----
    // (overlaps with the in-flight TDM transfer of this chunk)
    {
      const long pb = ((long)b * N_DIM + n0 + nn) * 2;
      const float px = pos[pb];
      const float py = pos[pb + 1];
#pragma unroll
      for (int i = 0; i < 8; ++i) {
        const int gr = grbase + 2 * i;
        const int gi = g0 + gr;
        const float cx = -1.0f + (float)(gi >> 6) * step;   // linspace 'ij'
        const float cy = -1.0f + (float)(gi & 63) * step;
        const float dx = cx - px;
        const float dy = cy - py;
        const float w = expf(-(dx * dx + dy * dy) * INV_BW);
        Wls[gr * LDW + nn] = w;
        atomicAdd(&rowsum[gr], w);   // ds_add_f32
      }
    }

    // wave 0: prefetch next chunk, then guarantee current chunk landed
    if (wave == 0) {
      if (c + 1 < NCHUNK) {
        tdm_load_proj_chunk(
            proj + pbase_b + (long)(c + 1) * PCHUNK_ELTS,
            (unsigned int)(uintptr_t)(void*)(Pls + ((c + 1) & 1) * PBUF));
        __builtin_amdgcn_s_wait_tensorcnt(1);   // chunk c complete
      } else {
        __builtin_amdgcn_s_wait_tensorcnt(0);   // last chunk complete
      }
    }
    __syncthreads();   // publish Wls + staged proj chunk to all waves

    // ---- phase B: acc += Wtile[16 x NKC] @ Pls[NKC x O-tiles] (all LDS) ----
    const float* Pb = Pls + (c & 1) * PBUF;
    for (int k = 0; k < NKC; k += 4) {
      const int kk = k + 2 * khalf;
      v2f a;
      a.x = Wls[mrow * LDW + kk];
      a.y = Wls[mrow * LDW + kk + 1];
      const int pr0 = kk * LDP;
      const int pr1 = pr0 + LDP;
      v2f bv0, bv1;
      bv0.x = Pb[pr0 + col0];
      bv0.y = Pb[pr1 + col0];
      acc0 = wmma_f32_4(a, bv0, acc0);
      bv1.x = Pb[pr0 + col1];
      bv1.y = Pb[pr1 + col1];
      acc1 = wmma_f32_4(a, bv1, acc1);
    }
    __syncthreads();   // done reading Wls and buffer (c&1) before reuse
  }

  // ---- normalize by (sum_n w + eps) and store ----
#pragma unroll
  for (int r = 0; r < 8; ++r) {
    const int M = r + 8 * khalf;
    const float s = 1.0f / (rowsum[M] + EPS_RBF);
    const long obase = (((long)b * G_DIM) + g0 + M) * O_DIM;
    out[obase + col0] = acc0[r] * s;
    out[obase + col1] = acc1[r] * s;
  }
}

// ---------------------------------------------------------------------------
// launcher
// ---------------------------------------------------------------------------
extern "C" void kernel_launch(void* const* d_in, const int* in_sizes, int n_in,
                              void* d_out, int out_size, void* d_ws, size_t ws_size,
                              hipStream_t stream) {
  const float* X   = (const float*)d_in[0];  // [B,N,D]
  const float* pos = (const float*)d_in[1];  // [B,N,2]
  const float* W1  = (const float*)d_in[2];  // [D,H]
  const float* b1  = (const float*)d_in[3];  // [H]
  const float* W2  = (const float*)d_in[4];  // [H,O]
  const float* b2  = (const float*)d_in[5];  // [O]
  float* out  = (float*)d_out;               // [B,G,O]
  float* proj = (float*)d_ws;                // [B*N, O] fp32 = 32 MB scratch

  // kernel 1: fused MLP -> projected
  dim3 blk(256);
  dim3 grd1((B_DIM * N_DIM) / 16);
  cog_mlp_kernel<<<grd1, blk, 0, stream>>>(X, W1, b1, W2, b2, proj);

  // kernel 2: TDM-staged, fused RBF weights + pooling GEMM + normalization
  const size_t smem2 = (size_t)(16 * LDW + 16 + 2 * PBUF) * sizeof(float); // ~272 KB
  (void)hipFuncSetAttribute((const void*)cog_pool_kernel,
                            hipFuncAttributeMaxDynamicSharedMemorySize,
                            (int)smem2);  // idempotent + deterministic
  dim3 grd2(G_DIM / 16, B_DIM);
  cog_pool_kernel<<<grd2, blk, smem2, stream>>>(pos, proj, out);
}